// RGAT_39273180955005
// MI455X (gfx1250) — compile-verified
//
#include <hip/hip_runtime.h>
#include <math.h>

#define N_NODES 20000
#define E_EDGES 320000
#define IN_CH   300
#define HID_CH  100
#define NB      8
#define BO      800   /* NB * HID_CH */
#define CLS_N   3
#define QN      64
#define EPSV    1e-16f
#define KCHUNK  32

typedef float v2f __attribute__((ext_vector_type(2)));
typedef float v8f __attribute__((ext_vector_type(8)));
typedef unsigned int u32x4 __attribute__((ext_vector_type(4)));
typedef int i32x4 __attribute__((ext_vector_type(4)));
typedef int i32x8 __attribute__((ext_vector_type(8)));
typedef int b128_t __attribute__((vector_size(16)));   // matches builtin's V4i param

// ---------------- gfx1250 data-movement helpers (all guarded) ----------------

// Async global->LDS copy of 16 bytes (tracked by ASYNCcnt).
__device__ __forceinline__ void async_copy_b128(const float* g, float* l) {
#if __has_builtin(__builtin_amdgcn_global_load_async_to_lds_b128)
  __builtin_amdgcn_global_load_async_to_lds_b128(
      (__attribute__((address_space(1))) b128_t*)g,
      (__attribute__((address_space(3))) b128_t*)l, 0, 0);
#else
  float4 t = *(const float4*)g;
  *(float4*)l = t;
#endif
}

__device__ __forceinline__ void wait_async0() {
#if __has_builtin(__builtin_amdgcn_s_wait_asynccnt)
  __builtin_amdgcn_s_wait_asynccnt(0);
#else
  asm volatile("s_wait_asynccnt 0x0" ::: "memory");
#endif
}

__device__ __forceinline__ void wait_tensor0() {
#if __has_builtin(__builtin_amdgcn_s_wait_tensorcnt)
  __builtin_amdgcn_s_wait_tensorcnt(0);
#else
  asm volatile("s_wait_tensorcnt 0x0" ::: "memory");
#endif
}

// TDM: DMA a 2D tile (tile_y rows x tile_x f32 elems, row stride = stride_elems)
// from global into contiguous LDS. D# packed per CDNA5 ISA 8.3/8.4.
__device__ __forceinline__ void tdm_load_2d(const float* gsrc, float* ldst,
                                            int tile_x, int tile_y,
                                            int stride_elems,
                                            int tensor_x, int tensor_y) {
#if __has_builtin(__builtin_amdgcn_tensor_load_to_lds)
  unsigned long long ga = (unsigned long long)gsrc;
  unsigned lds_off =
      (unsigned)(unsigned long long)(__attribute__((address_space(3))) float*)ldst;
  u32x4 g0;
  g0[0] = 1u;                                       // count=1 (valid user D#)
  g0[1] = lds_off;                                  // lds_addr (bytes)
  g0[2] = (unsigned)(ga & 0xffffffffu);             // global_addr[31:0]
  g0[3] = (unsigned)((ga >> 32) & 0x1ffffffu)       // global_addr[56:32]
          | (2u << 30);                             // type=2 ("image")
  i32x8 g1;
  g1[0] = (2 << 16);                                // wg_mask=0, data_size=2 (4B)
  g1[1] = (tensor_x & 0xffff) << 16;                // tensor_dim0[15:0] @ bit48
  g1[2] = ((tensor_x >> 16) & 0xffff)               // tensor_dim0[31:16]
          | ((tensor_y & 0xffff) << 16);            // tensor_dim1[15:0]
  g1[3] = ((tensor_y >> 16) & 0xffff)               // tensor_dim1[31:16]
          | ((tile_x & 0xffff) << 16);              // tile_dim0
  g1[4] = (tile_y & 0xffff);                        // tile_dim1 (tile_dim2=0)
  g1[5] = stride_elems;                             // tensor_dim0_stride[31:0]
  g1[6] = 0;                                        // stride[47:32], dim1_stride lo
  g1[7] = 0;
  i32x4 gz = {0, 0, 0, 0};
#if __clang_major__ >= 23
  i32x8 gz8 = {0, 0, 0, 0, 0, 0, 0, 0};
  __builtin_amdgcn_tensor_load_to_lds(g0, g1, gz, gz, gz8, 0);
#else
  __builtin_amdgcn_tensor_load_to_lds(g0, g1, gz, gz, 0);
#endif
#else
  // fallback: the wave copies the tile directly
  for (int i = threadIdx.x; i < tile_x * tile_y; i += 32) {
    int r = i / tile_x, c = i - r * tile_x;
    ldst[i] = gsrc[(size_t)r * stride_elems + c];
  }
#endif
}

// ---------- order-preserving float <-> uint for atomic max ----------
__device__ __forceinline__ unsigned f2ord(float f) {
  unsigned u = __float_as_uint(f);
  return (u & 0x80000000u) ? ~u : (u | 0x80000000u);
}
__device__ __forceinline__ float ord2f(unsigned u) {
  return (u & 0x80000000u) ? __uint_as_float(u & 0x7fffffffu) : __uint_as_float(~u);
}

// ---------- pad activations: xpad[n,kpad] = (k<cin) ? xin[n,k] : 0 ----------
__global__ void k_pad(const float* __restrict__ xin, float* __restrict__ xpad,
                      int cin, int kpad) {
  int i = blockIdx.x * blockDim.x + threadIdx.x;
  if (i >= N_NODES * kpad) return;
  int n = i / kpad, k = i - n * kpad;
  xpad[i] = (k < cin) ? xin[(size_t)n * cin + k] : 0.f;
}

// ---------- repack basis [B, cin, HID] -> Wmat [kpad, B*HID], zero rows >= cin ----------
__global__ void k_repack(const float* __restrict__ basis, float* __restrict__ Wmat,
                         int cin, int kpad) {
  int i = blockIdx.x * blockDim.x + threadIdx.x;
  int tot = kpad * BO;
  if (i >= tot) return;
  int row = i / BO, col = i - row * BO;
  int b = col / HID_CH, o = col - b * HID_CH;
  Wmat[i] = (row < cin) ? basis[((size_t)b * cin + row) * HID_CH + o] : 0.f;
}

// ---------- xb = Xpad[N,Kpad] @ Wmat[Kpad,BO] via V_WMMA_F32_16X16X4_F32 ----------
// One wave per 16x32 output tile. A tile staged via async global->LDS (b128),
// B tile staged via TDM tensor_load_to_lds; fragments consumed from LDS.
// A frag (16x4 f32, v2f/lane): lanes 0-15 -> K=k,k+1 (M=lane); lanes 16-31 -> K=k+2,k+3.
// B frag (4x16 f32, v2f/lane): VGPR0 = row K (lanes 0-15) / K+2 (lanes 16-31); VGPR1 = K+1.
// C/D (v8f): element r -> row M = r + 8*(lane>>4), col N = lane&15.
__global__ __launch_bounds__(32) void k_gemm_wmma(const float* __restrict__ A,
                                                  const float* __restrict__ Bm,
                                                  float* __restrict__ C, int Kpad) {
  __shared__ float ldsA[16 * KCHUNK];      // 2 KB
  __shared__ float ldsB[KCHUNK * 32];      // 4 KB
  const int lane = threadIdx.x;
  const int half = lane >> 4;
  const int l    = lane & 15;
  const int tm   = blockIdx.y << 4;        // 16 output rows
  const int tn   = blockIdx.x << 5;        // 32 output cols
  const int arow = lane >> 3;              // 0..3   (A staging pattern)
  const int ac4  = (lane & 7) << 2;        // 0,4,..,28
  v8f acc0 = {};
  v8f acc1 = {};

  for (int k0 = 0; k0 < Kpad; k0 += KCHUNK) {
    // stage A tile 16x32: 4 async b128 per lane
    #pragma unroll
    for (int j = 0; j < 4; ++j) {
      int r = arow + j * 4;
      async_copy_b128(A + (size_t)(tm + r) * Kpad + k0 + ac4,
                      &ldsA[r * KCHUNK + ac4]);
    }
    // stage B tile 32x32 via Tensor Data Mover
    tdm_load_2d(Bm + (size_t)k0 * BO + tn, &ldsB[0],
                /*tile_x=*/32, /*tile_y=*/KCHUNK,
                /*stride=*/BO, /*tensor_x=*/BO, /*tensor_y=*/Kpad);
    wait_async0();
    wait_tensor0();

    #pragma unroll
    for (int kk = 0; kk < KCHUNK; kk += 4) {
      const int ka = kk + 2 * half;
      v2f a;
      a.x = ldsA[l * KCHUNK + ka];
      a.y = ldsA[l * KCHUNK + ka + 1];
      v2f bA, bB;
      bA.x = ldsB[ka * 32 + l];
      bA.y = ldsB[(ka + 1) * 32 + l];
      bB.x = ldsB[ka * 32 + 16 + l];
      bB.y = ldsB[(ka + 1) * 32 + 16 + l];
#if __has_builtin(__builtin_amdgcn_wmma_f32_16x16x4_f32)
      acc0 = __builtin_amdgcn_wmma_f32_16x16x4_f32(false, a, false, bA, (short)0, acc0, false, false);
      acc1 = __builtin_amdgcn_wmma_f32_16x16x4_f32(false, a, false, bB, (short)0, acc1, false, false);
#else
      #pragma unroll
      for (int r = 0; r < 8; ++r) {
        acc0[r] += a.x * bA.x + a.y * bA.y;
        acc1[r] += a.x * bB.x + a.y * bB.y;
      }
#endif
    }
  }
  const int cm = tm + 8 * half;
  #pragma unroll
  for (int r = 0; r < 8; ++r) {
    C[(size_t)(cm + r) * BO + tn + l]      = acc0[r];
    C[(size_t)(cm + r) * BO + tn + 16 + l] = acc1[r];
  }
}

// ---------- u[n,b] = xb[n,b,:].q ;  v[n,b] = xb[n,b,:].k ----------
__global__ void k_uv(const float* __restrict__ xb, const float* __restrict__ qv,
                     const float* __restrict__ kv, float* __restrict__ u,
                     float* __restrict__ v) {
  int i = blockIdx.x * blockDim.x + threadIdx.x;
  if (i >= N_NODES * NB) return;
  int n = i / NB, b = i - n * NB;
  const float* __restrict__ p = xb + (size_t)n * BO + b * HID_CH;
  float su = 0.f, sv = 0.f;
  #pragma unroll 4
  for (int o = 0; o < HID_CH; ++o) {
    float xv = p[o];
    su += xv * qv[o];
    sv += xv * kv[o];
  }
  u[i] = su;
  v[i] = sv;
}

// ---------- zero/neg-inf init for softmax + aggregation buffers ----------
__global__ void k_init(unsigned* __restrict__ mord, float* __restrict__ ssum,
                       float* __restrict__ agg) {
  int i = blockIdx.x * blockDim.x + threadIdx.x;
  if (i < N_NODES) {
    mord[i] = 0x00800000u;  // f2ord(-FLT_MAX)
    ssum[i] = 0.f;
  }
  if (i < N_NODES * HID_CH) agg[i] = 0.f;
}

// ---------- edge pass 1: raw attention logit + segment max ----------
__global__ void k_edge_alpha(const int* __restrict__ ei, const int* __restrict__ etype,
                             const float* __restrict__ comp, const float* __restrict__ u,
                             const float* __restrict__ v, float* __restrict__ araw,
                             unsigned* __restrict__ mord) {
  int e = blockIdx.x * blockDim.x + threadIdx.x;
  if (e >= E_EDGES) return;
  int s = ei[e], d = ei[E_EDGES + e], r = etype[e];
  const float* __restrict__ cr = comp + r * NB;
  const float* __restrict__ ud = u + (size_t)d * NB;
  const float* __restrict__ vs = v + (size_t)s * NB;
  float a = 0.f;
  #pragma unroll
  for (int b = 0; b < NB; ++b) a += cr[b] * (ud[b] + vs[b]);  // qi + kj
  a = (a > 0.f) ? a : 0.2f * a;                               // leaky_relu(.,0.2)
  araw[e] = a;
  atomicMax(&mord[d], f2ord(a));
}

// ---------- edge pass 2: exp(a - max) + segment sum ----------
__global__ void k_edge_exp(const int* __restrict__ ei, const float* __restrict__ araw,
                           const unsigned* __restrict__ mord, float* __restrict__ eexp,
                           float* __restrict__ ssum) {
  int e = blockIdx.x * blockDim.x + threadIdx.x;
  if (e >= E_EDGES) return;
  int d = ei[E_EDGES + e];
  float m = ord2f(mord[d]);
  float ex = expf(araw[e] - m);
  eexp[e] = ex;
  atomicAdd(&ssum[d], ex);
}

// ---------- edge pass 3: agg[dst,o] += w_e * sum_b comp[et,b]*xb[src,b,o] ----------
__global__ void k_edge_scatter(const int* __restrict__ ei, const int* __restrict__ etype,
                               const float* __restrict__ comp, const float* __restrict__ xb,
                               const float* __restrict__ eexp, const float* __restrict__ ssum,
                               float* __restrict__ agg) {
  int i = blockIdx.x * blockDim.x + threadIdx.x;
  if (i >= E_EDGES * HID_CH) return;   // 32M threads: fits int32
  int e = i / HID_CH, o = i - e * HID_CH;
  int s = ei[e], d = ei[E_EDGES + e], r = etype[e];
  float w = eexp[e] / (ssum[d] + EPSV);
  const float* __restrict__ cr = comp + r * NB;
  const float* __restrict__ px = xb + (size_t)s * BO + o;
  __builtin_prefetch(px, 0, 1);
  float acc = 0.f;
  #pragma unroll
  for (int b = 0; b < NB; ++b) acc += cr[b] * px[b * HID_CH];
  atomicAdd(&agg[(size_t)d * HID_CH + o], w * acc);
}

// ---------- epilogue: h = relu(agg + bias) ----------
__global__ void k_epilogue(const float* __restrict__ agg, const float* __restrict__ bias,
                           float* __restrict__ hout) {
  int i = blockIdx.x * blockDim.x + threadIdx.x;
  if (i >= N_NODES * HID_CH) return;
  float t = agg[i] + bias[i % HID_CH];
  hout[i] = (t > 0.f) ? t : 0.f;
}

// ---------- final: mean-pool query rows, tiny linear head ----------
__global__ void k_final(const float* __restrict__ h2, const int* __restrict__ qidx,
                        const float* __restrict__ Wl, const float* __restrict__ bl,
                        float* __restrict__ out) {
  __shared__ float pooled[HID_CH];
  int t = threadIdx.x;
  if (t < HID_CH) {
    float s = 0.f;
    for (int q = 0; q < QN; ++q) s += h2[(size_t)qidx[q] * HID_CH + t];
    pooled[t] = s * (1.0f / QN);
  }
  __syncthreads();
  if (t < CLS_N) {
    float s = bl[t];
    for (int o = 0; o < HID_CH; ++o) s += pooled[o] * Wl[t * HID_CH + o];
    out[t] = s;
  }
}

// ---------- one RGAT layer (host-side launch sequence) ----------
static void run_layer(const float* xin, int cin, const float* comp, const float* basis,
                      const float* qv, const float* kv, const float* bias,
                      const int* ei, const int* etype,
                      float* xpad, float* Wmat, float* xb, float* u, float* v,
                      unsigned* mord, float* ssum, float* araw, float* eexp,
                      float* agg, float* hout, hipStream_t stream) {
  const int kpad = (cin + KCHUNK - 1) / KCHUNK * KCHUNK;  // 320 or 128
  k_pad<<<(N_NODES * kpad + 255) / 256, 256, 0, stream>>>(xin, xpad, cin, kpad);
  k_repack<<<(kpad * BO + 255) / 256, 256, 0, stream>>>(basis, Wmat, cin, kpad);
  dim3 g(BO / 32, N_NODES / 16);
  k_gemm_wmma<<<g, 32, 0, stream>>>(xpad, Wmat, xb, kpad);
  k_uv<<<(N_NODES * NB + 255) / 256, 256, 0, stream>>>(xb, qv, kv, u, v);
  k_init<<<(N_NODES * HID_CH + 255) / 256, 256, 0, stream>>>(mord, ssum, agg);
  k_edge_alpha<<<(E_EDGES + 255) / 256, 256, 0, stream>>>(ei, etype, comp, u, v, araw, mord);
  k_edge_exp<<<(E_EDGES + 255) / 256, 256, 0, stream>>>(ei, araw, mord, eexp, ssum);
  int tot3 = E_EDGES * HID_CH;  // 32,000,000
  k_edge_scatter<<<(tot3 + 255) / 256, 256, 0, stream>>>(ei, etype, comp, xb, eexp, ssum, agg);
  k_epilogue<<<(N_NODES * HID_CH + 255) / 256, 256, 0, stream>>>(agg, bias, hout);
}

extern "C" void kernel_launch(void* const* d_in, const int* in_sizes, int n_in,
                              void* d_out, int out_size, void* d_ws, size_t ws_size,
                              hipStream_t stream) {
  (void)in_sizes; (void)n_in; (void)out_size; (void)ws_size;
  const float* x      = (const float*)d_in[0];
  const int*   ei     = (const int*)d_in[1];
  const int*   etype  = (const int*)d_in[2];
  const int*   qidx   = (const int*)d_in[3];
  const float* comp1  = (const float*)d_in[4];
  const float* basis1 = (const float*)d_in[5];
  const float* q1     = (const float*)d_in[6];
  const float* k1     = (const float*)d_in[7];
  const float* b1     = (const float*)d_in[8];
  const float* comp2  = (const float*)d_in[9];
  const float* basis2 = (const float*)d_in[10];
  const float* q2     = (const float*)d_in[11];
  const float* k2     = (const float*)d_in[12];
  const float* b2     = (const float*)d_in[13];
  const float* Wl     = (const float*)d_in[14];
  const float* bl     = (const float*)d_in[15];

  char* ws = (char*)d_ws;
  size_t off = 0;
  auto take = [&](size_t bytes) -> void* {
    void* p = ws + off;
    off += (bytes + 255) & ~(size_t)255;
    return p;
  };
  float*    xpad = (float*)take((size_t)N_NODES * 320 * 4);      // padded activations
  float*    Wmat = (float*)take((size_t)320 * BO * 4);           // padded weights
  float*    xb   = (float*)take((size_t)N_NODES * BO * 4);       // 64 MB, L2-resident
  float*    u    = (float*)take((size_t)N_NODES * NB * 4);
  float*    v    = (float*)take((size_t)N_NODES * NB * 4);
  unsigned* mord = (unsigned*)take((size_t)N_NODES * 4);
  float*    ssum = (float*)take((size_t)N_NODES * 4);
  float*    araw = (float*)take((size_t)E_EDGES * 4);
  float*    eexp = (float*)take((size_t)E_EDGES * 4);
  float*    agg  = (float*)take((size_t)N_NODES * HID_CH * 4);
  float*    h1   = (float*)take((size_t)N_NODES * HID_CH * 4);
  float*    h2   = (float*)take((size_t)N_NODES * HID_CH * 4);

  run_layer(x, IN_CH, comp1, basis1, q1, k1, b1, ei, etype,
            xpad, Wmat, xb, u, v, mord, ssum, araw, eexp, agg, h1, stream);
  run_layer(h1, HID_CH, comp2, basis2, q2, k2, b2, ei, etype,
            xpad, Wmat, xb, u, v, mord, ssum, araw, eexp, agg, h2, stream);
  k_final<<<1, 128, 0, stream>>>(h2, qidx, Wl, bl, (float*)d_out);
}